// GaussianModel_57191784513783
// MI455X (gfx1250) — compile-verified
//
#include <hip/hip_runtime.h>
#include <cstdint>
#include <cstddef>

#define TILE 256

// ---- CDNA5 async global<->LDS helpers (gfx1250, tracked by ASYNCcnt) ----

// Truncate a generic pointer to a shared-memory object down to the 32-bit
// wave-relative LDS byte offset (the low 32 bits of the generic address are the LDS offset).
__device__ __forceinline__ unsigned lds_off(const void* p) {
    return (unsigned)(unsigned long long)p;
}

// Copies 16 bytes from the global address gaddr to the LDS address lds_addr, per active lane.
__device__ __forceinline__ void async_load_b128(unsigned lds_addr, const void* gaddr) {
    asm volatile("global_load_async_to_lds_b128 %0, %1, off"
                 :
                 : "v"(lds_addr), "v"(gaddr)
                 : "memory");
}

// Copies 16 bytes from the LDS address lds_addr to the global address gaddr, per active lane.
__device__ __forceinline__ void async_store_b128(void* gaddr, unsigned lds_addr) {
    asm volatile("global_store_async_from_lds_b128 %0, %1, off"
                 :
                 : "v"(gaddr), "v"(lds_addr)
                 : "memory");
}

__device__ __forceinline__ void wait_async0() {
    asm volatile("s_wait_asynccnt 0x0" ::: "memory");
}

// ---- kernel: per-Gaussian covariance  Sigma = (R*diag(exp(s))) (R*diag(exp(s)))^T ----

__global__ __launch_bounds__(TILE)
void GaussianModel_cov_kernel(const float* __restrict__ scale,
                              const float* __restrict__ rot,
                              float* __restrict__ out, int n) {
    __shared__ __align__(16) float s_rot[TILE * 4];    // 4096 B
    __shared__ __align__(16) float s_scale[TILE * 3];  // 3072 B
    __shared__ __align__(16) float s_out[TILE * 9];    // 9216 B

    const int t = threadIdx.x;
    const int base = blockIdx.x * TILE;
    const bool full = (base + TILE) <= n;   // uniform per block

    if (full) {
        // rot tile: 256 * 16 B, one b128 per lane, perfectly coalesced.
        async_load_b128(lds_off(&s_rot[t * 4]),
                        rot + (size_t)(base + t) * 4);
        // scale tile: 256*3 floats = 192 * 16 B chunks.
        if (t < (TILE * 3) / 4) {
            async_load_b128(lds_off(&s_scale[t * 4]),
                            scale + (size_t)base * 3 + (size_t)t * 4);
        }
        wait_async0();          // this wave's async loads are in LDS
    } else {
        const int g = base + t;
        if (g < n) {
            s_rot[t * 4 + 0] = rot[(size_t)g * 4 + 0];
            s_rot[t * 4 + 1] = rot[(size_t)g * 4 + 1];
            s_rot[t * 4 + 2] = rot[(size_t)g * 4 + 2];
            s_rot[t * 4 + 3] = rot[(size_t)g * 4 + 3];
            s_scale[t * 3 + 0] = scale[(size_t)g * 3 + 0];
            s_scale[t * 3 + 1] = scale[(size_t)g * 3 + 1];
            s_scale[t * 3 + 2] = scale[(size_t)g * 3 + 2];
        }
    }
    __syncthreads();            // cross-wave: all tiles staged

    if (base + t < n) {
        const float4 q4 = *(const float4*)&s_rot[t * 4];   // ds_load_b128
        float qr = q4.x, qi = q4.y, qj = q4.z, qk = q4.w;

        // normalize quaternion (match reference: max(norm, 1e-12))
        const float n2  = qr * qr + qi * qi + qj * qj + qk * qk;
        const float nrm = fmaxf(sqrtf(n2), 1e-12f);
        const float inv = 1.0f / nrm;
        qr *= inv; qi *= inv; qj *= inv; qk *= inv;

        // stride-3 LDS reads: gcd(3,64)=1 -> bank-conflict-free across a wave
        const float s0 = __expf(s_scale[t * 3 + 0]);
        const float s1 = __expf(s_scale[t * 3 + 1]);
        const float s2 = __expf(s_scale[t * 3 + 2]);

        const float r00 = 1.0f - 2.0f * (qj * qj + qk * qk);
        const float r01 = 2.0f * (qi * qj - qk * qr);
        const float r02 = 2.0f * (qi * qk + qj * qr);
        const float r10 = 2.0f * (qi * qj + qk * qr);
        const float r11 = 1.0f - 2.0f * (qi * qi + qk * qk);
        const float r12 = 2.0f * (qj * qk - qi * qr);
        const float r20 = 2.0f * (qi * qk - qj * qr);
        const float r21 = 2.0f * (qj * qk + qi * qr);
        const float r22 = 1.0f - 2.0f * (qi * qi + qj * qj);

        // M = R * diag(s)
        const float m00 = r00 * s0, m01 = r01 * s1, m02 = r02 * s2;
        const float m10 = r10 * s0, m11 = r11 * s1, m12 = r12 * s2;
        const float m20 = r20 * s0, m21 = r21 * s1, m22 = r22 * s2;

        // Sigma = M * M^T (symmetric)
        const float c00 = m00 * m00 + m01 * m01 + m02 * m02;
        const float c01 = m00 * m10 + m01 * m11 + m02 * m12;
        const float c02 = m00 * m20 + m01 * m21 + m02 * m22;
        const float c11 = m10 * m10 + m11 * m11 + m12 * m12;
        const float c12 = m10 * m20 + m11 * m21 + m12 * m22;
        const float c22 = m20 * m20 + m21 * m21 + m22 * m22;

        // stride-9 LDS writes: gcd(9,64)=1 -> conflict-free
        float* o = &s_out[t * 9];
        o[0] = c00; o[1] = c01; o[2] = c02;
        o[3] = c01; o[4] = c11; o[5] = c12;
        o[6] = c02; o[7] = c12; o[8] = c22;
    }
    __syncthreads();            // tile's outputs staged in LDS

    if (full) {
        // 256*9 floats = 576 x 16 B chunks; each wave's b128 store covers
        // 512 contiguous bytes = 4 fully-written 128B cachelines.
        float* ob = out + (size_t)base * 9;
        async_store_b128(ob + (size_t)t * 4,              lds_off(&s_out[t * 4]));
        async_store_b128(ob + (size_t)(TILE + t) * 4,     lds_off(&s_out[(TILE + t) * 4]));
        if (t < (TILE * 9) / 4 - 2 * TILE) {              // remaining 64 chunks
            async_store_b128(ob + (size_t)(2 * TILE + t) * 4,
                             lds_off(&s_out[(2 * TILE + t) * 4]));
        }
        wait_async0();          // S_ENDPGM also implies wait-idle; explicit for clarity
    } else {
        const int g = base + t;
        if (g < n) {
            #pragma unroll
            for (int k = 0; k < 9; ++k)
                out[(size_t)g * 9 + k] = s_out[t * 9 + k];
        }
    }
}

extern "C" void kernel_launch(void* const* d_in, const int* in_sizes, int n_in,
                              void* d_out, int out_size, void* d_ws, size_t ws_size,
                              hipStream_t stream) {
    const float* scale = (const float*)d_in[0];   // (N,3) f32
    const float* rot   = (const float*)d_in[1];   // (N,4) f32
    float*       out   = (float*)d_out;           // (N,3,3) f32

    const int n = in_sizes[0] / 3;                // N = 4,000,000 -> 15625 full tiles
    const int blocks = (n + TILE - 1) / TILE;

    hipLaunchKernelGGL(GaussianModel_cov_kernel,
                       dim3(blocks), dim3(TILE), 0, stream,
                       scale, rot, out, n);
}